// GCN_2903397892205
// MI455X (gfx1250) — compile-verified
//
#include <hip/hip_runtime.h>
#include <hip/hip_bf16.h>

// ---------------------------------------------------------------------------
// GCN (3x GCNConv + BN + ReLU, mean-pool, 2-layer head) for MI455X (gfx1250).
// Memory-bound workload (edge gather/scatter ~5-6 GB total traffic); GEMMs are
// done in full f32 precision with V_WMMA_F32_16X16X4_F32 (wave32 WMMA).
// ---------------------------------------------------------------------------

static constexpr int HD  = 128;   // hidden dim
static constexpr int NG  = 64;    // number of graphs
static constexpr int ODIM = 16;   // output dim
static constexpr float BN_EPS = 1e-5f;

typedef __attribute__((ext_vector_type(2))) float v2f;
typedef __attribute__((ext_vector_type(8))) float v8f;

// ---------------------------------------------------------------------------
// degree / normalization
// ---------------------------------------------------------------------------
__global__ void deg_init_kernel(float* deg, int n) {
    int i = blockIdx.x * blockDim.x + threadIdx.x;
    if (i < n) deg[i] = 1.0f;                      // self-loop contributes 1
}

__global__ void deg_scatter_kernel(const int* __restrict__ dst, float* deg, int e) {
    int i = blockIdx.x * blockDim.x + threadIdx.x;
    if (i < e) unsafeAtomicAdd(&deg[dst[i]], 1.0f);
}

__global__ void deg_rsqrt_kernel(float* deg, int n) {
    int i = blockIdx.x * blockDim.x + threadIdx.x;
    if (i < n) deg[i] = rsqrtf(deg[i]);            // deg >= 1 always
}

// ---------------------------------------------------------------------------
// layer 0 linear: h = x (N,1) * W0 (1,H)
// ---------------------------------------------------------------------------
__global__ void lin0_kernel(const float* __restrict__ x, const float* __restrict__ W0,
                            float* __restrict__ hlin, int n) {
    size_t i = (size_t)blockIdx.x * blockDim.x + threadIdx.x;
    if (i >= (size_t)n * HD) return;
    int row = (int)(i >> 7);
    int c   = (int)(i & (HD - 1));
    hlin[i] = x[row] * W0[c];
}

// ---------------------------------------------------------------------------
// WMMA f32 GEMM: C(M,Ncols) = A(M,128) * B(128,Ncols) [+bias][ReLU]
// One wave computes one 16x16 tile via 32x V_WMMA_F32_16X16X4_F32.
// A layout (16x4 f32): lanes 0-15 row M, v0=K0 v1=K1; lanes 16-31 v0=K2 v1=K3.
// B layout (4x16 f32): lanes hold col N; lanes 0-15 v0=K0 v1=K1; 16-31 K2/K3.
// C/D layout (16x16 f32): VGPR j, lanes 0-15 -> M=j, lanes 16-31 -> M=j+8.
// EXEC is all-ones within every live wave (tile guard is wave-uniform).
// ---------------------------------------------------------------------------
__global__ void gemm_wmma_kernel(const float* __restrict__ A, const float* __restrict__ B,
                                 float* __restrict__ C, int M, int Ncols,
                                 const float* __restrict__ bias, int relu) {
    const int lane = threadIdx.x & 31;
    const int wave = threadIdx.x >> 5;
    const int tile = blockIdx.x * (blockDim.x >> 5) + wave;
    const int ntn  = Ncols >> 4;
    const int tm   = tile / ntn;
    const int tn   = tile - tm * ntn;
    if (tm * 16 >= M) return;                      // wave-uniform exit

    const int mrow = tm * 16 + (lane & 15);
    const int ncol = tn * 16 + (lane & 15);
    const int kh   = (lane >> 4) * 2;              // 0 or 2

    const float* ap = A + (size_t)mrow * HD + kh;
    const float* bp = B + (size_t)kh * Ncols + ncol;
    __builtin_prefetch(ap, 0, 1);                  // global_prefetch_b8

    v8f acc = {};
#pragma unroll 4
    for (int k0 = 0; k0 < HD; k0 += 4) {
        v2f a = *(const v2f*)(ap + k0);            // contiguous K pair
        v2f b;
        b.x = bp[(size_t)k0 * Ncols];
        b.y = bp[(size_t)(k0 + 1) * Ncols];
        acc = __builtin_amdgcn_wmma_f32_16x16x4_f32(
            /*neg_a=*/false, a, /*neg_b=*/false, b,
            /*c_mod=*/(short)0, acc, /*reuse_a=*/false, /*reuse_b=*/false);
    }

    const float bv = bias ? bias[ncol] : 0.0f;
    const int rb = tm * 16 + (lane >> 4) * 8;
#pragma unroll
    for (int j = 0; j < 8; ++j) {
        float v = acc[j] + bv;
        if (relu) v = fmaxf(v, 0.0f);
        C[(size_t)(rb + j) * Ncols + ncol] = v;
    }
}

// ---------------------------------------------------------------------------
// agg = bias + dis[i]^2 * hlin[i]   (self-loop term, norm = dis*dis)
// ---------------------------------------------------------------------------
__global__ void agg_init_kernel(const float* __restrict__ hlin, const float* __restrict__ dis,
                                const float* __restrict__ b, float* __restrict__ agg, int n) {
    size_t i = (size_t)blockIdx.x * blockDim.x + threadIdx.x;
    if (i >= (size_t)n * HD) return;
    int row = (int)(i >> 7);
    int c   = (int)(i & (HD - 1));
    float d = dis[row];
    agg[i] = b[c] + d * d * hlin[i];
}

// ---------------------------------------------------------------------------
// edge scatter: agg[dst] += dis[src]*dis[dst] * hlin[src]
// block = (32,8): one wave per edge, each lane handles 4 channels (float4).
// ---------------------------------------------------------------------------
__global__ void edge_scatter_kernel(const int* __restrict__ src, const int* __restrict__ dst,
                                    const float* __restrict__ dis, const float* __restrict__ h,
                                    float* __restrict__ agg, int e) {
    int ed = blockIdx.x * blockDim.y + threadIdx.y;
    if (ed >= e) return;
    int s = src[ed];
    int d = dst[ed];
    float w = dis[s] * dis[d];
    int c = threadIdx.x * 4;
    const float4 hv = *(const float4*)(h + (size_t)s * HD + c);
    float* ap = agg + (size_t)d * HD + c;
    unsafeAtomicAdd(ap + 0, w * hv.x);
    unsafeAtomicAdd(ap + 1, w * hv.y);
    unsafeAtomicAdd(ap + 2, w * hv.z);
    unsafeAtomicAdd(ap + 3, w * hv.w);
}

// ---------------------------------------------------------------------------
// BN stats: per-channel sum / sumsq over N rows (block = 128 ch, 512 rows)
// ---------------------------------------------------------------------------
__global__ void bn_stats_kernel(const float* __restrict__ a, float* sums, int n) {
    const int c = threadIdx.x;                     // 0..127
    int r0 = blockIdx.x * 512;
    int r1 = min(n, r0 + 512);
    float s = 0.0f, ss = 0.0f;
    for (int r = r0; r < r1; ++r) {
        float v = a[(size_t)r * HD + c];
        s += v;
        ss += v * v;
    }
    unsafeAtomicAdd(&sums[c], s);
    unsafeAtomicAdd(&sums[HD + c], ss);
}

__global__ void bn_apply_kernel(float* __restrict__ a, const float* __restrict__ sums,
                                const float* __restrict__ g, const float* __restrict__ be, int n) {
    size_t i = (size_t)blockIdx.x * blockDim.x + threadIdx.x;
    if (i >= (size_t)n * HD) return;
    int c = (int)(i & (HD - 1));
    float inv_n = 1.0f / (float)n;
    float mean = sums[c] * inv_n;
    float var  = sums[HD + c] * inv_n - mean * mean;
    float v = (a[i] - mean) * rsqrtf(var + BN_EPS) * g[c] + be[c];
    a[i] = fmaxf(v, 0.0f);
}

// ---------------------------------------------------------------------------
// pooling: counts via LDS histogram; sums via run-length compressed atomics
// (batch is sorted, so register accumulation flushes only at graph changes)
// ---------------------------------------------------------------------------
__global__ void count_kernel(const int* __restrict__ batch, float* counts, int n) {
    __shared__ float lc[NG];
    if (threadIdx.x < NG) lc[threadIdx.x] = 0.0f;
    __syncthreads();
    int i = blockIdx.x * blockDim.x + threadIdx.x;
    if (i < n) atomicAdd(&lc[batch[i]], 1.0f);     // ds_add_f32
    __syncthreads();
    if (threadIdx.x < NG) {
        float v = lc[threadIdx.x];
        if (v != 0.0f) unsafeAtomicAdd(&counts[threadIdx.x], v);
    }
}

__global__ void pool_accum_kernel(const float* __restrict__ h, const int* __restrict__ batch,
                                  float* pooled, int n) {
    const int c = threadIdx.x;                     // 0..127
    int r0 = blockIdx.x * 256;
    if (r0 >= n) return;
    int r1 = min(n, r0 + 256);
    float acc = 0.0f;
    int cur = batch[r0];
    for (int r = r0; r < r1; ++r) {
        int b = batch[r];
        if (b != cur) {
            unsafeAtomicAdd(&pooled[cur * HD + c], acc);
            acc = 0.0f;
            cur = b;
        }
        acc += h[(size_t)r * HD + c];
    }
    unsafeAtomicAdd(&pooled[cur * HD + c], acc);
}

__global__ void pool_div_kernel(float* pooled, const float* counts) {
    int i = blockIdx.x * blockDim.x + threadIdx.x;
    if (i >= NG * HD) return;
    pooled[i] /= fmaxf(counts[i >> 7], 1.0f);
}

// ---------------------------------------------------------------------------
static inline int cdiv(int a, int b) { return (a + b - 1) / b; }

extern "C" void kernel_launch(void* const* d_in, const int* in_sizes, int n_in,
                              void* d_out, int out_size, void* d_ws, size_t ws_size,
                              hipStream_t stream) {
    // inputs in setup_inputs() order (JAX x64 disabled -> int32 indices)
    const float* x    = (const float*)d_in[0];
    const int*   ei   = (const int*)d_in[1];     // (2, E)
    const int*   bat  = (const int*)d_in[2];
    const float* W0   = (const float*)d_in[3];
    const float* b0   = (const float*)d_in[4];
    const float* g0   = (const float*)d_in[5];
    const float* be0  = (const float*)d_in[6];
    const float* W1   = (const float*)d_in[7];
    const float* b1   = (const float*)d_in[8];
    const float* g1   = (const float*)d_in[9];
    const float* be1  = (const float*)d_in[10];
    const float* W2   = (const float*)d_in[11];
    const float* b2   = (const float*)d_in[12];
    const float* g2   = (const float*)d_in[13];
    const float* be2  = (const float*)d_in[14];
    const float* Wl   = (const float*)d_in[15];
    const float* bl   = (const float*)d_in[16];
    const float* Wo   = (const float*)d_in[17];
    const float* bo   = (const float*)d_in[18];

    const int N = in_sizes[0];
    const int E = in_sizes[1] / 2;
    const int* srcI = ei;          // edge_index[0]
    const int* dstI = ei + E;      // edge_index[1]

    // workspace layout (floats)
    float* ws     = (float*)d_ws;
    float* dis    = ws;  ws += N;                  // degree, then rsqrt(degree)
    float* hlin   = ws;  ws += (size_t)N * HD;     // linear output
    float* agg    = ws;  ws += (size_t)N * HD;     // aggregated / activated
    float* sums   = ws;  ws += 2 * HD;             // BN sum / sumsq
    float* pooled = ws;  ws += NG * HD;
    float* counts = ws;  ws += NG;
    float* hidden = ws;  ws += NG * HD;
    float* out    = (float*)d_out;                 // (64, 16)

    // --- gcn_norm: deg (with self loop) -> dis = rsqrt(deg) ----------------
    deg_init_kernel<<<cdiv(N, 256), 256, 0, stream>>>(dis, N);
    deg_scatter_kernel<<<cdiv(E, 256), 256, 0, stream>>>(dstI, dis, E);
    deg_rsqrt_kernel<<<cdiv(N, 256), 256, 0, stream>>>(dis, N);

    // --- layer 0 linear (1 -> 128) -----------------------------------------
    lin0_kernel<<<cdiv(N * HD, 256), 256, 0, stream>>>(x, W0, hlin, N);

    const float* Wv[3]  = {nullptr, W1, W2};
    const float* bv[3]  = {b0, b1, b2};
    const float* gv[3]  = {g0, g1, g2};
    const float* bev[3] = {be0, be1, be2};

    const dim3 eblk(32, 8);
    for (int L = 0; L < 3; ++L) {
        if (L > 0) {
            const int tiles = cdiv(N, 16) * (HD / 16);
            gemm_wmma_kernel<<<cdiv(tiles, 8), 256, 0, stream>>>(
                agg, Wv[L], hlin, N, HD, nullptr, 0);
        }
        hipMemsetAsync(sums, 0, 2 * HD * sizeof(float), stream);
        agg_init_kernel<<<cdiv(N * HD, 256), 256, 0, stream>>>(hlin, dis, bv[L], agg, N);
        edge_scatter_kernel<<<cdiv(E, 8), eblk, 0, stream>>>(srcI, dstI, dis, hlin, agg, E);
        bn_stats_kernel<<<cdiv(N, 512), HD, 0, stream>>>(agg, sums, N);
        bn_apply_kernel<<<cdiv(N * HD, 256), 256, 0, stream>>>(agg, sums, gv[L], bev[L], N);
    }

    // --- global mean pool ---------------------------------------------------
    hipMemsetAsync(pooled, 0, NG * HD * sizeof(float), stream);
    hipMemsetAsync(counts, 0, NG * sizeof(float), stream);
    count_kernel<<<cdiv(N, 256), 256, 0, stream>>>(bat, counts, N);
    pool_accum_kernel<<<cdiv(N, 256), HD, 0, stream>>>(agg, bat, pooled, N);
    pool_div_kernel<<<cdiv(NG * HD, 256), 256, 0, stream>>>(pooled, counts);

    // --- head: relu(pooled @ Wl + bl) @ Wo + bo ----------------------------
    {
        const int tiles1 = (NG / 16) * (HD / 16);  // 32
        gemm_wmma_kernel<<<cdiv(tiles1, 8), 256, 0, stream>>>(
            pooled, Wl, hidden, NG, HD, bl, 1);
        const int tiles2 = (NG / 16) * (ODIM / 16); // 4
        gemm_wmma_kernel<<<cdiv(tiles2, 8), 256, 0, stream>>>(
            hidden, Wo, out, NG, ODIM, bo, 0);
    }
}